// GraphSAGE_1082331758914
// MI455X (gfx1250) — compile-verified
//
#include <hip/hip_runtime.h>

typedef __attribute__((ext_vector_type(2))) float v2f;
typedef __attribute__((ext_vector_type(8))) float v8f;

#define NNODES 50000
#define TSTRIDE 130   // LDS tile row stride (floats): 130 % 64 == 2 -> conflict-free b64 frag reads

// ---------------------------------------------------------------- zero fill
__global__ void zero_f32(float* __restrict__ p, long long n) {
  long long i = (long long)blockIdx.x * blockDim.x + threadIdx.x;
  long long stride = (long long)gridDim.x * blockDim.x;
  for (; i < n; i += stride) p[i] = 0.0f;
}

// ---------------------------------------------------------------- in-degree
__global__ void degree_count(const int* __restrict__ dst, int E,
                             float* __restrict__ deg) {
  int e = blockIdx.x * blockDim.x + threadIdx.x;
  if (e < E) unsafeAtomicAdd(&deg[dst[e]], 1.0f);
}

// ------------------------------------------------- scatter-add (segment sum)
// One thread = one edge x 4 consecutive features (float4 gather + 4 HW f32
// atomics into L2). chunks = d/4 is a power of two (logc).
__global__ void scatter_add4(const float* __restrict__ X,
                             const int* __restrict__ src,
                             const int* __restrict__ dst,
                             int E, int d, int logc,
                             float* __restrict__ agg) {
  long long gid = (long long)blockIdx.x * blockDim.x + threadIdx.x;
  int chunks = 1 << logc;
  if (gid >= (long long)E * chunks) return;
  int e = (int)(gid >> logc);
  int c = ((int)gid & (chunks - 1)) << 2;
  int s = src[e], t = dst[e];
  const float4 v = *(const float4*)(X + (size_t)s * d + c);
  float* o = agg + (size_t)t * d + c;
  unsafeAtomicAdd(o + 0, v.x);
  unsafeAtomicAdd(o + 1, v.y);
  unsafeAtomicAdd(o + 2, v.z);
  unsafeAtomicAdd(o + 3, v.w);
}

// ------------------------------------------------- row scale by 1/max(deg,1)
__global__ void scale_by_invdeg(float* __restrict__ agg,
                                const float* __restrict__ deg,
                                long long nvec, int logd2) {
  long long gid = (long long)blockIdx.x * blockDim.x + threadIdx.x;
  if (gid >= nvec) return;
  int row = (int)(gid >> logd2);
  float s = 1.0f / fmaxf(deg[row], 1.0f);
  float4* p = (float4*)agg + gid;
  float4 v = *p;
  v.x *= s; v.y *= s; v.z *= s; v.w *= s;
  *p = v;
}

// ------------------------------------------- fp32 WMMA dual-GEMM, LDS-staged B
// C = A@W [+ A2@W2] [+ D] [+ bias] [relu] via V_WMMA_F32_16X16X4_F32.
// A:[M,K] rm, W:[K,N] rm, C/D:[M,N]. One wave32 computes a 16x32 output strip
// (two 16x16 WMMA tiles sharing the A fragment); 8 waves/block -> 128 rows x
// 32 cols per block. W slice for the block's 32 columns is staged once in LDS,
// transposed [n][k] with row stride 130 floats so the v2f (ds_load_b64)
// fragment reads are bank-conflict-free. Requires M%16==0 (50000=3125*16),
// K%4==0, N%32==0. All threads reach __syncthreads(); compute is guarded by
// the wave-uniform row check so EXEC is all-ones inside WMMA.
__launch_bounds__(256)
__global__ void gemm_wmma_f32(const float* __restrict__ A,
                              const float* __restrict__ W,
                              const float* __restrict__ A2,
                              const float* __restrict__ W2,
                              const float* __restrict__ D,
                              const float* __restrict__ bias,
                              float* __restrict__ C,
                              int M, int K, int N, int relu) {
  extern __shared__ float smem[];
  float* tT0 = smem;                     // [32][TSTRIDE]
  float* tT1 = smem + 32 * TSTRIDE;      // second tile iff A2 != null

  const int tid  = threadIdx.x;
  const int col0 = blockIdx.y * 32;

  // cooperative transpose-fill: tT[n][k] = W[k][col0+n]
  for (int idx = tid; idx < K * 8; idx += 256) {
    int k = idx >> 3, c4 = (idx & 7) << 2;
    const float4 v = *(const float4*)(W + (size_t)k * N + col0 + c4);
    tT0[(c4 + 0) * TSTRIDE + k] = v.x;
    tT0[(c4 + 1) * TSTRIDE + k] = v.y;
    tT0[(c4 + 2) * TSTRIDE + k] = v.z;
    tT0[(c4 + 3) * TSTRIDE + k] = v.w;
  }
  if (A2 != nullptr) {
    for (int idx = tid; idx < K * 8; idx += 256) {
      int k = idx >> 3, c4 = (idx & 7) << 2;
      const float4 v = *(const float4*)(W2 + (size_t)k * N + col0 + c4);
      tT1[(c4 + 0) * TSTRIDE + k] = v.x;
      tT1[(c4 + 1) * TSTRIDE + k] = v.y;
      tT1[(c4 + 2) * TSTRIDE + k] = v.z;
      tT1[(c4 + 3) * TSTRIDE + k] = v.w;
    }
  }
  __syncthreads();

  const int lane = tid & 31;
  const int wave = tid >> 5;
  const int row0 = (blockIdx.x * 8 + wave) * 16;
  const int half = lane >> 4;            // lanes 16-31 carry K=2,3 (ISA 16x4 layout)
  const int l    = lane & 15;

  if (row0 < M) {                        // wave-uniform
    v8f acc0 = {0.f, 0.f, 0.f, 0.f, 0.f, 0.f, 0.f, 0.f};
    v8f acc1 = {0.f, 0.f, 0.f, 0.f, 0.f, 0.f, 0.f, 0.f};

    {
      const float* Ar = A + (size_t)(row0 + l) * K + 2 * half;
      const float* B0 = tT0 + l * TSTRIDE + 2 * half;
      const float* B1 = tT0 + (l + 16) * TSTRIDE + 2 * half;
#pragma unroll 8
      for (int k = 0; k < K; k += 4) {
        v2f a;
        a.x = Ar[k];
        a.y = Ar[k + 1];
        v2f b0 = *(const v2f*)(B0 + k);
        v2f b1 = *(const v2f*)(B1 + k);
        acc0 = __builtin_amdgcn_wmma_f32_16x16x4_f32(false, a, false, b0,
                                                     (short)0, acc0, false, false);
        acc1 = __builtin_amdgcn_wmma_f32_16x16x4_f32(false, a, false, b1,
                                                     (short)0, acc1, false, false);
      }
    }
    if (A2 != nullptr) {                 // fused second GEMM into same tiles
      const float* Ar = A2 + (size_t)(row0 + l) * K + 2 * half;
      const float* B0 = tT1 + l * TSTRIDE + 2 * half;
      const float* B1 = tT1 + (l + 16) * TSTRIDE + 2 * half;
#pragma unroll 8
      for (int k = 0; k < K; k += 4) {
        v2f a;
        a.x = Ar[k];
        a.y = Ar[k + 1];
        v2f b0 = *(const v2f*)(B0 + k);
        v2f b1 = *(const v2f*)(B1 + k);
        acc0 = __builtin_amdgcn_wmma_f32_16x16x4_f32(false, a, false, b0,
                                                     (short)0, acc0, false, false);
        acc1 = __builtin_amdgcn_wmma_f32_16x16x4_f32(false, a, false, b1,
                                                     (short)0, acc1, false, false);
      }
    }

    const int n0 = col0 + l;
    const int n1 = n0 + 16;
    const float bi0 = bias ? bias[n0] : 0.0f;
    const float bi1 = bias ? bias[n1] : 0.0f;
#pragma unroll
    for (int v = 0; v < 8; ++v) {        // D layout: row = v + 8*half, col = lane&15
      int r = row0 + v + 8 * half;
      float v0 = acc0[v] + bi0;
      float v1 = acc1[v] + bi1;
      if (D) {
        v0 += D[(size_t)r * N + n0];
        v1 += D[(size_t)r * N + n1];
      }
      if (relu) {
        v0 = fmaxf(v0, 0.0f);
        v1 = fmaxf(v1, 0.0f);
      }
      C[(size_t)r * N + n0] = v0;
      C[(size_t)r * N + n1] = v1;
    }
  }
}

// ------------------------------------------------------------------ launcher
extern "C" void kernel_launch(void* const* d_in, const int* in_sizes, int n_in,
                              void* d_out, int out_size, void* d_ws, size_t ws_size,
                              hipStream_t stream) {
  const float* x   = (const float*)d_in[0];
  const int*   ei  = (const int*)d_in[1];     // [2, E] row-major
  const float* W1l = (const float*)d_in[2];
  const float* W1r = (const float*)d_in[3];
  const float* b1  = (const float*)d_in[4];
  const float* W2l = (const float*)d_in[5];
  const float* W2r = (const float*)d_in[6];
  const float* b2  = (const float*)d_in[7];

  const int E  = in_sizes[1] / 2;
  const int* src = ei;
  const int* dst = ei + E;
  const int N = NNODES, DH = 128, DO = 64;

  const size_t smem1 = 1 * 32 * TSTRIDE * sizeof(float);   // single GEMM
  const size_t smem2 = 2 * 32 * TSTRIDE * sizeof(float);   // fused dual GEMM

  // workspace layout (floats): deg | agg1 | h | p | agg2   (~77 MB)
  float* deg  = (float*)d_ws;                       // N (padded to 50176)
  float* agg1 = deg  + 50176;                       // N*128
  float* h    = agg1 + (size_t)N * DH;              // N*128
  float* p    = h    + (size_t)N * DH;              // N*64
  float* agg2 = p    + (size_t)N * DO;              // N*64
  float* out  = (float*)d_out;                      // N*64

  zero_f32<<<2048, 256, 0, stream>>>(deg, 50176LL + (long long)N * DH);
  zero_f32<<<1024, 256, 0, stream>>>(agg2, (long long)N * DO);

  degree_count<<<(E + 255) / 256, 256, 0, stream>>>(dst, E, deg);

  // ---- layer 1: scatter-mean on raw features (d=128) --------------------
  {
    long long work = (long long)E * 32;             // E * d/4
    scatter_add4<<<(int)((work + 255) / 256), 256, 0, stream>>>(
        x, src, dst, E, 128, 5, agg1);
  }
  scale_by_invdeg<<<(N * 32 + 255) / 256, 256, 0, stream>>>(
      agg1, deg, (long long)N * 32, 5);

  // h = relu(agg1 @ W1l + x @ W1r + b1)
  {
    dim3 g((N + 127) / 128, DH / 32);
    gemm_wmma_f32<<<g, 256, smem2, stream>>>(agg1, W1l, x, W1r,
                                             nullptr, b1, h, N, 128, DH, 1);
  }

  // ---- layer 2: project BEFORE aggregating (mean & linear commute) ------
  // p = h @ W2l  (128 -> 64) halves the scatter volume.
  {
    dim3 g((N + 127) / 128, DO / 32);
    gemm_wmma_f32<<<g, 256, smem1, stream>>>(h, W2l, nullptr, nullptr,
                                             nullptr, nullptr, p, N, 128, DO, 0);
  }
  {
    long long work = (long long)E * 16;             // E * 64/4
    scatter_add4<<<(int)((work + 255) / 256), 256, 0, stream>>>(
        p, src, dst, E, 64, 4, agg2);
  }
  scale_by_invdeg<<<(N * 16 + 255) / 256, 256, 0, stream>>>(
      agg2, deg, (long long)N * 16, 4);

  // out = h @ W2r + agg2 + b2
  {
    dim3 g((N + 127) / 128, DO / 32);
    gemm_wmma_f32<<<g, 256, smem1, stream>>>(h, W2r, nullptr, nullptr,
                                             agg2, b2, out, N, 128, DO, 0);
  }
}